// ChainGraphDQN_45019847197224
// MI455X (gfx1250) — compile-verified
//
#include <hip/hip_runtime.h>
#include <stdint.h>

// Problem constants (from reference)
#define NN   200000   // nodes
#define FIN  128      // in features
#define EE   3200000  // edges
#define GG   4096     // graphs
#define HH   16       // GCN out channels
#define DD1  64       // MLP hidden
#define MM   12       // microservices
#define AA   8        // actions

#define GEMM_WAVES 4  // waves (tiles) per block in the WMMA GEMM

typedef __attribute__((ext_vector_type(2))) float v2f;
typedef __attribute__((ext_vector_type(8))) float v8f;

// ---------------------------------------------------------------------------
// K1: degree count over edge destinations (self-loop +1 folded in later)
// ---------------------------------------------------------------------------
__global__ void k_degree(const int* __restrict__ dst, float* __restrict__ deg) {
    int e = blockIdx.x * blockDim.x + threadIdx.x;
    if (e < EE) atomicAdd(&deg[dst[e]], 1.0f);
}

// ---------------------------------------------------------------------------
// K2: deg -> dinv = rsqrt(deg_edges + 1)  (self loop guarantees deg>0)
// ---------------------------------------------------------------------------
__global__ void k_dinv(float* __restrict__ deg) {
    int i = blockIdx.x * blockDim.x + threadIdx.x;
    if (i < NN) deg[i] = rsqrtf(deg[i] + 1.0f);
}

// ---------------------------------------------------------------------------
// K3: hs = (x @ conv_w) * dinv[row]   [N,128]x[128,16]
// V_WMMA_F32_16X16X4_F32, one 16-row tile per wave, 32 K-steps.
// A-tile (16x128 fp32, 8 KB) staged to LDS with GLOBAL_LOAD_ASYNC_TO_LDS_B128
// (coalesced 16B/lane copies, ASYNCcnt-tracked), fenced by s_wait_asynccnt.
// Epilogue folds the dinv[s] factor of the symmetric GCN norm into h.
// ---------------------------------------------------------------------------
__global__ void k_gemm_xw(const float* __restrict__ x,
                          const float* __restrict__ w,
                          const float* __restrict__ dinv,
                          float* __restrict__ hs) {
    __shared__ float sA[GEMM_WAVES][16 * FIN];   // 4 x 8 KB = 32 KB
    const int lane = threadIdx.x & 31;
    const int wave = threadIdx.x >> 5;
    const int tile = blockIdx.x * GEMM_WAVES + wave;   // 12500 tiles (N/16)
    if (tile >= (NN / 16)) return;                     // wave-uniform guard

    // ---- async stage: 8192 B tile = 16 iterations x 32 lanes x 16 B ----
    {
        const uint64_t gbase = (uint64_t)(uintptr_t)(x + (size_t)tile * 16 * FIN);
        const uint32_t lbase = (uint32_t)(uintptr_t)(&sA[wave][0]);
#pragma unroll
        for (int c = 0; c < 16; ++c) {
            const uint32_t byteoff = (uint32_t)(c * 32 + lane) * 16u;
            uint32_t ldst  = lbase + byteoff;
            uint64_t gaddr = gbase + byteoff;
            asm volatile("global_load_async_to_lds_b128 %0, %1, off"
                         :: "v"(ldst), "v"(gaddr) : "memory");
        }
        asm volatile("s_wait_asynccnt 0" ::: "memory");
    }

    const int half = lane >> 4;            // 0 or 1
    const int mn   = lane & 15;            // M for A, N for B/D
    const float* ar = &sA[wave][mn * FIN];

    v8f c = {};
#pragma unroll
    for (int k0 = 0; k0 < FIN; k0 += 4) {
        const int kk = k0 + 2 * half;
        v2f a, b;
        a.x = ar[kk];
        a.y = ar[kk + 1];
        b.x = w[kk * HH + mn];
        b.y = w[(kk + 1) * HH + mn];
        c = __builtin_amdgcn_wmma_f32_16x16x4_f32(
                /*neg_a=*/false, a, /*neg_b=*/false, b,
                /*c_mod=*/(short)0, c, /*reuse_a=*/false, /*reuse_b=*/false);
    }
#pragma unroll
    for (int v = 0; v < 8; ++v) {
        const int r = tile * 16 + v + 8 * half;
        hs[(size_t)r * HH + mn] = c[v] * dinv[r];   // fold dinv[src] factor
    }
}

// ---------------------------------------------------------------------------
// K4: edge scatter with factorized norm: agg[d] += hs[s]
// (dinv[d] applied once per node in K5). 16 lanes per edge -> 64B coalesced
// atomic bursts into the L2-resident agg buffer (12.8 MB << 192 MB L2).
// global_prefetch hints cover the 25.6 MB streaming edge-index reads.
// ---------------------------------------------------------------------------
__global__ void k_edges(const int* __restrict__ ei,
                        const float* __restrict__ hs,
                        float* __restrict__ agg) {
    int idx = blockIdx.x * blockDim.x + threadIdx.x;   // EE*16 = 51.2M < 2^31
    if (idx >= EE * HH) return;
    int e = idx >> 4;
    int f = idx & 15;
    if (f == 0 && e + 4096 < EE) {
        __builtin_prefetch(ei + e + 4096, 0, 0);           // src stream
        __builtin_prefetch(ei + EE + e + 4096, 0, 0);      // dst stream
    }
    int s = ei[e];
    int d = ei[EE + e];
    atomicAdd(&agg[(size_t)d * HH + f], hs[(size_t)s * HH + f]);
}

// ---------------------------------------------------------------------------
// K5: fused dinv[d] scale + self-loop + bias + ReLU + mean-pool accumulation.
//   h2[i] = relu(dinv[i]*(agg[i] + hs[i]) + conv_b)   (self loop = dinv*hs)
// h2 never materialized; pooled/cnt are tiny (L2-resident) atomics.
// ---------------------------------------------------------------------------
__global__ void k_pool(const float* __restrict__ agg,
                       const float* __restrict__ hs,
                       const float* __restrict__ dinv,
                       const float* __restrict__ conv_b,
                       const int* __restrict__ batch,
                       float* __restrict__ pooled,
                       float* __restrict__ cnt) {
    int idx = blockIdx.x * blockDim.x + threadIdx.x;
    if (idx >= NN * HH) return;
    int i = idx >> 4;
    int f = idx & 15;
    float v = dinv[i] * (agg[idx] + hs[idx]) + conv_b[f];
    v = fmaxf(v, 0.0f);                       // relu
    int b = batch[i];
    atomicAdd(&pooled[b * HH + f], v);
    if (f == 0) atomicAdd(&cnt[b], 1.0f);
}

// ---------------------------------------------------------------------------
// K6: mean normalize + MLP(elu) x2 + per-microservice heads.
// One thread per graph; all weights staged in LDS (~45 KB of 320 KB/WGP).
// ---------------------------------------------------------------------------
__global__ void k_mlp_heads(const float* __restrict__ pooled,
                            const float* __restrict__ cnt,
                            const float* __restrict__ w1, const float* __restrict__ b1,
                            const float* __restrict__ w2, const float* __restrict__ b2,
                            const float* __restrict__ hw, const float* __restrict__ hb,
                            float* __restrict__ out) {
    __shared__ float s_w1[HH * DD1];
    __shared__ float s_w2[DD1 * DD1];
    __shared__ float s_hw[MM * DD1 * AA];
    __shared__ float s_b1[DD1];
    __shared__ float s_b2[DD1];
    __shared__ float s_hb[MM * AA];

    for (int i = threadIdx.x; i < HH * DD1; i += blockDim.x)       s_w1[i] = w1[i];
    for (int i = threadIdx.x; i < DD1 * DD1; i += blockDim.x)      s_w2[i] = w2[i];
    for (int i = threadIdx.x; i < MM * DD1 * AA; i += blockDim.x)  s_hw[i] = hw[i];
    if (threadIdx.x < DD1) { s_b1[threadIdx.x] = b1[threadIdx.x];
                             s_b2[threadIdx.x] = b2[threadIdx.x]; }
    if (threadIdx.x < MM * AA) s_hb[threadIdx.x] = hb[threadIdx.x];
    __syncthreads();

    int g = blockIdx.x * blockDim.x + threadIdx.x;
    if (g >= GG) return;

    float inv = 1.0f / fmaxf(cnt[g], 1.0f);
    float p[HH];
#pragma unroll
    for (int k = 0; k < HH; ++k) p[k] = pooled[g * HH + k] * inv;

    float d1[DD1];
#pragma unroll 4
    for (int j = 0; j < DD1; ++j) {
        float acc = s_b1[j];
#pragma unroll
        for (int k = 0; k < HH; ++k) acc = fmaf(p[k], s_w1[k * DD1 + j], acc);
        d1[j] = (acc > 0.0f) ? acc : (expf(acc) - 1.0f);   // elu
    }

    float d2[DD1];
#pragma unroll 2
    for (int j = 0; j < DD1; ++j) {
        float acc = s_b2[j];
#pragma unroll 8
        for (int k = 0; k < DD1; ++k) acc = fmaf(d1[k], s_w2[k * DD1 + j], acc);
        d2[j] = (acc > 0.0f) ? acc : (expf(acc) - 1.0f);   // elu
    }

    for (int m = 0; m < MM; ++m) {
        const float* wm = &s_hw[m * DD1 * AA];
        float acc[AA];
#pragma unroll
        for (int a = 0; a < AA; ++a) acc[a] = s_hb[m * AA + a];
        for (int k = 0; k < DD1; ++k) {
            float dk = d2[k];
#pragma unroll
            for (int a = 0; a < AA; ++a) acc[a] = fmaf(dk, wm[k * AA + a], acc[a]);
        }
#pragma unroll
        for (int a = 0; a < AA; ++a) out[(size_t)g * (MM * AA) + m * AA + a] = acc[a];
    }
}

// ---------------------------------------------------------------------------
extern "C" void kernel_launch(void* const* d_in, const int* in_sizes, int n_in,
                              void* d_out, int out_size, void* d_ws, size_t ws_size,
                              hipStream_t stream) {
    const float* x      = (const float*)d_in[0];
    const int*   ei     = (const int*)d_in[1];     // [2,E] flat: src @0, dst @E
    const int*   batch  = (const int*)d_in[2];
    const float* conv_w = (const float*)d_in[3];
    const float* conv_b = (const float*)d_in[4];
    const float* w1     = (const float*)d_in[5];
    const float* b1     = (const float*)d_in[6];
    const float* w2     = (const float*)d_in[7];
    const float* b2     = (const float*)d_in[8];
    const float* head_w = (const float*)d_in[9];
    const float* head_b = (const float*)d_in[10];
    float* out = (float*)d_out;

    // Workspace layout (floats)
    float* hs     = (float*)d_ws;                      // N*16  (h * dinv)
    float* dinv   = hs + (size_t)NN * HH;              // N   (deg, then in-place rsqrt)
    float* agg    = dinv + NN;                         // N*16
    float* pooled = agg + (size_t)NN * HH;             // G*16
    float* cnt    = pooled + (size_t)GG * HH;          // G

    // Zero deg/agg/pooled/cnt (contiguous region after hs)
    size_t zero_bytes = ((size_t)NN + (size_t)NN * HH + (size_t)GG * HH + GG) * sizeof(float);
    hipMemsetAsync(dinv, 0, zero_bytes, stream);

    const int B = 256;
    k_degree<<<(EE + B - 1) / B, B, 0, stream>>>(ei + EE, dinv);
    k_dinv  <<<(NN + B - 1) / B, B, 0, stream>>>(dinv);
    k_gemm_xw<<<(NN / 16 + GEMM_WAVES - 1) / GEMM_WAVES, GEMM_WAVES * 32, 0, stream>>>(
        x, conv_w, dinv, hs);
    k_edges <<<((EE * HH) + B - 1) / B, B, 0, stream>>>(ei, hs, agg);
    k_pool  <<<((NN * HH) + B - 1) / B, B, 0, stream>>>(agg, hs, dinv, conv_b, batch,
                                                        pooled, cnt);
    k_mlp_heads<<<(GG + B - 1) / B, B, 0, stream>>>(pooled, cnt, w1, b1, w2, b2,
                                                    head_w, head_b, out);
}